// InterMeanLoss_27341761806532
// MI455X (gfx1250) — compile-verified
//
#include <hip/hip_runtime.h>
#include <hip/hip_bf16.h>

typedef float v2f __attribute__((ext_vector_type(2)));
typedef float v8f __attribute__((ext_vector_type(8)));

#define CDIM 128
#define DDIM 256
#define DT   64      // D-tile per segment-sum block
#define LPAD 65      // LDS row stride (pad to spread banks)

// ---------------------------------------------------------------- zero scratch
__global__ void iml_zero_kernel(float* p, int n) {
    int i = blockIdx.x * 256 + threadIdx.x;
    if (i < n) p[i] = 0.0f;
}

// ---------------------------------------------------------------- argmax rows
// one wave32 per row of logits[N][128]; float4 per lane; first-index tie-break
__global__ void iml_argmax_kernel(const float* __restrict__ logits,
                                  int* __restrict__ labels,
                                  float* __restrict__ counts, int N) {
    int row  = blockIdx.x * 8 + (threadIdx.x >> 5);
    int lane = threadIdx.x & 31;
    if (row >= N) return;

    const float4 v = ((const float4*)(logits + (size_t)row * CDIM))[lane];
    float best = v.x; int bi = lane * 4;
    if (v.y > best) { best = v.y; bi = lane * 4 + 1; }
    if (v.z > best) { best = v.z; bi = lane * 4 + 2; }
    if (v.w > best) { best = v.w; bi = lane * 4 + 3; }

    for (int off = 16; off > 0; off >>= 1) {
        float ov = __shfl_down(best, off, 32);
        int   oi = __shfl_down(bi,   off, 32);
        if (ov > best || (ov == best && oi < bi)) { best = ov; bi = oi; }
    }
    if (lane == 0) {
        labels[row] = bi;
        atomicAdd(&counts[bi], 1.0f);   // global_atomic_add_f32
    }
}

// ---------------------------------------------------------------- segment sum
// grid = (rowChunks, D/DT). LDS-privatized [128][DT] partial sums, ds_add_f32,
// then global f32 atomic flush.
__global__ void iml_segsum_kernel(const float* __restrict__ feats,
                                  const int* __restrict__ labels,
                                  float* __restrict__ sums,
                                  int rowsPerBlock) {
    __shared__ float lsum[CDIM * LPAD];
    for (int i = threadIdx.x; i < CDIM * LPAD; i += 256) lsum[i] = 0.0f;
    __syncthreads();

    const int dbase = blockIdx.y * DT;
    const int rbase = blockIdx.x * rowsPerBlock;
    const int c4    = (threadIdx.x & 15) * 4;   // 16 lanes cover 64 cols (float4)
    const int rofs  = threadIdx.x >> 4;         // 16 rows in flight per iter

    for (int r = rbase + rofs; r < rbase + rowsPerBlock; r += 16) {
        const int lab = labels[r];
        const float4 v = *(const float4*)(feats + (size_t)r * DDIM + dbase + c4);
        float* dst = &lsum[lab * LPAD + c4];
        atomicAdd(dst + 0, v.x);                // ds_add_f32
        atomicAdd(dst + 1, v.y);
        atomicAdd(dst + 2, v.z);
        atomicAdd(dst + 3, v.w);
    }
    __syncthreads();

    for (int i = threadIdx.x; i < CDIM * DT; i += 256) {
        const int lab = i >> 6;                 // /DT
        const int col = i & (DT - 1);
        const float val = lsum[lab * LPAD + col];
        if (val != 0.0f)
            atomicAdd(&sums[lab * DDIM + dbase + col], val);
    }
}

// ---------------------------------------------------------------- mean + L2 norm
// one block per class; 256 threads == D
__global__ void iml_nmean_kernel(const float* __restrict__ sums,
                                 const float* __restrict__ counts,
                                 float* __restrict__ nmeans) {
    const int c = blockIdx.x;
    const int t = threadIdx.x;
    const float cnt  = counts[c];
    const float mean = sums[c * DDIM + t] / fmaxf(cnt, 1.0f);

    __shared__ float red[256];
    red[t] = mean * mean;
    __syncthreads();
    for (int s = 128; s > 0; s >>= 1) {
        if (t < s) red[t] += red[t + s];
        __syncthreads();
    }
    const float norm  = sqrtf(red[0]);
    const float scale = norm > 0.0f ? 1.0f / norm : 1.0f;
    nmeans[c * DDIM + t] = mean * scale;
}

// ---------------------------------------------------------------- sim + loss
// 1 block, 8 waves. Wave w owns output rows [16w,16w+16); 8 tiles of 16x16,
// each accumulated over K=256 in steps of 4 with V_WMMA_F32_16X16X4_F32.
// f32 A 16x4 layout: lanes 0-15 -> M=lane, K={k,k+1}; lanes 16-31 -> K={k+2,k+3}.
// f32 B 4x16 layout mirrors it with N=lane. C/D: VGPR r -> M=r (+8 for hi half).
__global__ void iml_simloss_kernel(const float* __restrict__ nmeans,
                                   const float* __restrict__ counts,
                                   float* __restrict__ out) {
    const int wave = threadIdx.x >> 5;
    const int lane = threadIdx.x & 31;
    const int half = lane >> 4;
    const int lid  = lane & 15;
    const int ti   = wave;                       // 8 waves -> 8 row-tiles

    float acc = 0.0f;
    for (int tj = 0; tj < 8; ++tj) {
        v8f c = {0.f, 0.f, 0.f, 0.f, 0.f, 0.f, 0.f, 0.f};
        const v2f* arow = (const v2f*)(nmeans + (size_t)(ti * 16 + lid) * DDIM);
        const v2f* brow = (const v2f*)(nmeans + (size_t)(tj * 16 + lid) * DDIM);
        for (int k = 0; k < DDIM; k += 4) {
            v2f a = arow[(k >> 1) + half];       // K = k+2*half, k+2*half+1
            v2f b = brow[(k >> 1) + half];
            c = __builtin_amdgcn_wmma_f32_16x16x4_f32(
                    false, a, false, b, (short)0, c, false, false);
        }
        const int   col  = tj * 16 + lid;
        const float pcol = counts[col];
        #pragma unroll
        for (int r = 0; r < 8; ++r) {
            const int row = ti * 16 + r + 8 * half;
            if (row != col && counts[row] > 0.0f && pcol > 0.0f)
                acc += 1.0f - c[r];
        }
    }

    __shared__ float red[256];
    red[threadIdx.x] = acc;
    __syncthreads();
    for (int s = 128; s > 0; s >>= 1) {
        if (threadIdx.x < s) red[threadIdx.x] += red[threadIdx.x + s];
        __syncthreads();
    }
    if (threadIdx.x == 0) out[0] = red[0];
}

// ---------------------------------------------------------------- launcher
extern "C" void kernel_launch(void* const* d_in, const int* in_sizes, int n_in,
                              void* d_out, int out_size, void* d_ws, size_t ws_size,
                              hipStream_t stream) {
    const float* logits = (const float*)d_in[0];
    const float* feats  = (const float*)d_in[1];
    const int N = in_sizes[0] / CDIM;            // 262144

    char*  ws     = (char*)d_ws;
    int*   labels = (int*)ws;                                   // N ints
    float* sums   = (float*)(ws + (size_t)N * sizeof(int));     // 128*256
    float* counts = sums + CDIM * DDIM;                         // 128
    float* nmeans = counts + CDIM;                              // 128*256

    // zero sums + counts (contiguous)
    const int nz = CDIM * DDIM + CDIM;
    iml_zero_kernel<<<(nz + 255) / 256, 256, 0, stream>>>(sums, nz);

    // pass 1: argmax labels + class counts (reads 128 MiB)
    iml_argmax_kernel<<<N / 8, 256, 0, stream>>>(logits, labels, counts, N);

    // pass 2: LDS-privatized segment sum (reads 256 MiB)
    const int rowChunks = 256;
    dim3 g(rowChunks, DDIM / DT);
    iml_segsum_kernel<<<g, 256, 0, stream>>>(feats, labels, sums, N / rowChunks);

    // finalize means + L2 normalize
    iml_nmean_kernel<<<CDIM, 256, 0, stream>>>(sums, counts, nmeans);

    // cosine-sim GEMM via f32 WMMA + masked off-diagonal loss reduction
    iml_simloss_kernel<<<1, 256, 0, stream>>>(nmeans, counts, (float*)d_out);
}